// ObjectCondensation_13417477832897
// MI455X (gfx1250) — compile-verified
//
#include <hip/hip_runtime.h>
#include <hip/hip_bf16.h>
#include <math.h>

typedef __attribute__((ext_vector_type(2))) float v2f;
typedef __attribute__((ext_vector_type(8))) float v8f;

#define Q_MIN_C   0.1f
#define BSE_C     0.001f
#define EPS_LSE_C 0.001f

// Fast hardware sqrt (single v_sqrt_f32, ~1 ulp) — plenty for a loss weight.
#if defined(__has_builtin)
#if __has_builtin(__builtin_amdgcn_sqrtf)
#define FAST_SQRTF(x) __builtin_amdgcn_sqrtf(x)
#endif
#endif
#ifndef FAST_SQRTF
#define FAST_SQRTF(x) sqrtf(x)
#endif

// ---------- helpers ----------
__device__ __forceinline__ float block_reduce_sum(float v, float* sbuf) {
  int t = threadIdx.x;
  __syncthreads();
  sbuf[t] = v;
  __syncthreads();
  for (int s = blockDim.x >> 1; s > 0; s >>= 1) {
    if (t < s) sbuf[t] += sbuf[t + s];
    __syncthreads();
  }
  float r = sbuf[0];
  __syncthreads();
  return r;
}

// ---------- kernel 0: zero output tail + accumulator scratch ----------
__global__ void k_zero(float* out_tail, int n_tail, float* acc, int n_acc) {
  int i = blockIdx.x * blockDim.x + threadIdx.x;
  if (i < n_tail) out_tail[i] = 0.f;
  if (i < n_acc)  acc[i] = 0.f;
}

// ---------- kernel 1: per-point beta_scale and |coords|^2 ----------
__global__ void k_point(const float* __restrict__ beta,
                        const float* __restrict__ coords,
                        int N, int C,
                        float* __restrict__ bs, float* __restrict__ nb) {
  int n = blockIdx.x * blockDim.x + threadIdx.x;
  if (n >= N) return;
  float b = beta[n] / (1.f + BSE_C);
  float a = atanhf(b);
  bs[n] = a * a + Q_MIN_C;
  float s = 0.f;
  const float* cr = coords + (size_t)n * C;
  for (int c = 0; c < C; ++c) { float x = cr[c]; s += x * x; }
  nb[n] = s;
}

// ---------- kernel 2: per-cluster argmax / attractive / LSE / correction ----------
__global__ void k_cluster(const float* __restrict__ beta,
                          const float* __restrict__ coords,
                          const int*   __restrict__ M,
                          const float* __restrict__ bs,
                          int C, int Mmax,
                          float* __restrict__ xa, float* __restrict__ na,
                          float* __restrict__ bsk, float* __restrict__ cntf,
                          float* __restrict__ sumpl, float* __restrict__ LVk,
                          float* __restrict__ pen, float* __restrict__ repcorr) {
  __shared__ float s_v[256];
  __shared__ int   s_i[256];
  __shared__ float s_red[256];
  __shared__ float s_xa[16];
  __shared__ float s_bsk;
  __shared__ float s_mb;
  int k = blockIdx.x;
  int t = threadIdx.x;
  const int* Mrow = M + (size_t)k * Mmax;

  float best = -1.f; int bestm = 0;
  float mb = 0.f, sb = 0.f, sp = 0.f, cnt = 0.f;
  for (int m = t; m < Mmax; m += blockDim.x) {
    int idx = Mrow[m];
    if (idx >= 0) {
      float v = bs[idx];
      if (v > best) { best = v; bestm = m; }
      float be = beta[idx];
      mb = fmaxf(mb, be);
      sb += be;
      sp += v - Q_MIN_C;
      cnt += 1.f;
    }
  }
  // argmax reduce (first occurrence on ties)
  s_v[t] = best; s_i[t] = bestm;
  __syncthreads();
  for (int s = blockDim.x >> 1; s > 0; s >>= 1) {
    if (t < s) {
      if (s_v[t + s] > s_v[t] ||
          (s_v[t + s] == s_v[t] && s_i[t + s] < s_i[t])) {
        s_v[t] = s_v[t + s]; s_i[t] = s_i[t + s];
      }
    }
    __syncthreads();
  }
  // max beta reduce
  s_red[t] = mb;
  __syncthreads();
  for (int s = blockDim.x >> 1; s > 0; s >>= 1) {
    if (t < s) s_red[t] = fmaxf(s_red[t], s_red[t + s]);
    __syncthreads();
  }
  if (t == 0) s_mb = fmaxf(s_red[0], 0.f);  // implicit zeros from padding
  __syncthreads();

  float sumb = block_reduce_sum(sb,  s_red);
  float sump = block_reduce_sum(sp,  s_red);
  float cnts = block_reduce_sum(cnt, s_red);

  if (t == 0) {
    int alpha = Mrow[s_i[0]];
    if (alpha < 0) alpha = 0;               // safety (never happens in practice)
    float bk = bs[alpha];
    s_bsk = bk;
    float nrm = 0.f;
    const float* cr = coords + (size_t)alpha * C;
    for (int c = 0; c < C; ++c) { float x = cr[c]; s_xa[c] = x; xa[(size_t)k * C + c] = x; nrm += x * x; }
    na[k]  = nrm;
    bsk[k] = bk;
    cntf[k] = cnts;
    sumpl[k] = sump;
  }
  __syncthreads();

  float mbv = s_mb;
  float bk  = s_bsk;
  float att = 0.f, rep = 0.f, lses = 0.f;
  for (int m = t; m < Mmax; m += blockDim.x) {
    int idx = Mrow[m];
    if (idx >= 0) {
      float dsq = 0.f;
      const float* cr = coords + (size_t)idx * C;
      for (int c = 0; c < C; ++c) { float d = s_xa[c] - cr[c]; dsq += d * d; }
      float bi = bs[idx];
      att += dsq * bi;
      float r = 1.f - FAST_SQRTF(dsq + 1e-6f);
      rep += (r > 0.f ? r : 0.f) * bi;
      lses += expf((beta[idx] - mbv) / EPS_LSE_C);
    } else {
      lses += expf((0.f - mbv) / EPS_LSE_C);  // padded zeros enter the LSE
    }
  }
  att  = block_reduce_sum(att,  s_red);
  rep  = block_reduce_sum(rep,  s_red);
  lses = block_reduce_sum(lses, s_red);
  if (t == 0) {
    LVk[k]     = bk * att / (cnts + 1e-6f);
    repcorr[k] = bk * rep;
    float lse  = mbv / EPS_LSE_C + logf(lses);
    float sc   = fminf(fmaxf(sumb, 0.f), 1.f);
    pen[k]     = 1.f - EPS_LSE_C * lse + (1.f - sc);
  }
}

// ---------- kernel 3: noise segment sums ----------
__global__ void k_noise(const float* __restrict__ beta,
                        const int* __restrict__ asso,
                        const int* __restrict__ rowi,
                        int N, float* __restrict__ noise) {
  int n = blockIdx.x * blockDim.x + threadIdx.x;
  if (n >= N) return;
  if (asso[n] < 0) atomicAdd(&noise[rowi[n]], beta[n]);
}

// ---------- kernel 4: dense repulsive term via V_WMMA_F32_16X16X4_F32 ----------
__global__ void k_rep(const float* __restrict__ coords,
                      const float* __restrict__ bs,
                      const float* __restrict__ nb,
                      const float* __restrict__ xa,
                      const float* __restrict__ na,
                      const float* __restrict__ bsk,
                      int C, int Kper, int Nper, int KT,
                      float* __restrict__ lrepraw) {
  __shared__ float s_red[16];
  int bt = blockIdx.y;
  int b  = bt / KT;
  int kt = bt % KT;
  int kbase = b * Kper + kt * 16;
  int kend  = b * Kper + Kper;
  int t = threadIdx.x;
  if (t < 16) s_red[t] = 0.f;
  __syncthreads();

  int lane = t & 31;
  int wave = t >> 5;
  int half = lane >> 4;     // 0: lanes 0-15, 1: lanes 16-31
  int l    = lane & 15;

  // A operand (x_a tile, 16x8 f32 split into two 16x4 halves).
  // ISA layout: lanes 0-15 hold row M=l dims {K0,K1}; lanes 16-31 dims {K2,K3}.
  int arow = kbase + l;
  bool avalid = (arow < kend);
  const float* xar = xa + (size_t)arow * C;
  v2f a0, a1;
  a0.x = avalid ? xar[half * 2 + 0] : 0.f;
  a0.y = avalid ? xar[half * 2 + 1] : 0.f;
  a1.x = avalid ? xar[4 + half * 2 + 0] : 0.f;
  a1.y = avalid ? xar[4 + half * 2 + 1] : 0.f;

  // D layout: VGPR r -> row (half*8 + r), column = l.
  float na_r[8], bk_r[8], acc[8];
  for (int r = 0; r < 8; ++r) {
    int kr = kbase + half * 8 + r;
    bool v = (kr < kend);
    na_r[r] = v ? na[kr]  : 0.f;
    bk_r[r] = v ? bsk[kr] : 0.f;
    acc[r] = 0.f;
  }

  int nchunks = (Nper + 15) / 16;
  int wstride = gridDim.x * (blockDim.x >> 5);
  int nbeg = b * Nper, nlim = nbeg + Nper;

  for (int c = blockIdx.x * (blockDim.x >> 5) + wave; c < nchunks; c += wstride) {
    int n = nbeg + c * 16 + l;
    bool nv = (n < nlim);
    const float* cr = coords + (size_t)n * C;
    v2f b0, b1;
    b0.x = nv ? cr[half * 2 + 0] : 0.f;
    b0.y = nv ? cr[half * 2 + 1] : 0.f;
    b1.x = nv ? cr[4 + half * 2 + 0] : 0.f;
    b1.y = nv ? cr[4 + half * 2 + 1] : 0.f;
    float nbv = nv ? nb[n] : 0.f;
    float bsn = nv ? bs[n] : 0.f;

    v8f cacc = {};
    cacc = __builtin_amdgcn_wmma_f32_16x16x4_f32(false, a0, false, b0,
                                                 (short)0, cacc, false, false);
    cacc = __builtin_amdgcn_wmma_f32_16x16x4_f32(false, a1, false, b1,
                                                 (short)0, cacc, false, false);
    for (int r = 0; r < 8; ++r) {
      float dsq = na_r[r] + nbv - 2.f * cacc[r];
      dsq = fmaxf(dsq, 0.f);
      float v = 1.f - FAST_SQRTF(dsq + 1e-6f);
      v = (v > 0.f) ? v : 0.f;
      acc[r] += v * bsn;
    }
  }

  // reduce columns within each 16-lane half
  for (int r = 0; r < 8; ++r) {
    float a = acc[r];
    a += __shfl_xor(a, 1, 32);
    a += __shfl_xor(a, 2, 32);
    a += __shfl_xor(a, 4, 32);
    a += __shfl_xor(a, 8, 32);
    acc[r] = a;
  }
  if (l == 0) {
    for (int r = 0; r < 8; ++r)
      atomicAdd(&s_red[half * 8 + r], acc[r] * bk_r[r]);
  }
  __syncthreads();
  if (t < 16) {
    int kr = kbase + t;
    if (kr < kend) atomicAdd(&lrepraw[kr], s_red[t]);
  }
}

// ---------- kernel 5: finalize scalars ----------
__global__ void k_final(const float* __restrict__ LVk,
                        const float* __restrict__ pen,
                        const float* __restrict__ repcorr,
                        const float* __restrict__ lrepraw,
                        const float* __restrict__ cntf,
                        const int*   __restrict__ kper,
                        const float* __restrict__ noise,
                        const int*   __restrict__ lengths,
                        int K, int B, int Nper,
                        float* __restrict__ out,
                        float* __restrict__ lvrepk) {
  __shared__ float s_red[256];
  int t = threadIdx.x;
  float sLV = 0.f, sLr = 0.f, sP = 0.f;
  for (int k = t; k < K; k += blockDim.x) {
    float Kk = (float)kper[k];
    float lr = (lrepraw[k] - repcorr[k]) / ((float)Nper - cntf[k] + 1e-6f);
    sLV += LVk[k] / Kk;
    sLr += lr / Kk;
    sP  += pen[k] / Kk;
    lvrepk[k] = LVk[k] + lr;
  }
  sLV = block_reduce_sum(sLV, s_red);
  sLr = block_reduce_sum(sLr, s_red);
  sP  = block_reduce_sum(sP,  s_red);
  if (t == 0) {
    float nz = 0.f;
    for (int bb = 0; bb < B; ++bb) nz += noise[bb] / (float)lengths[bb];
    out[0] = sLV;
    out[1] = sLr;
    out[2] = sP + nz / (float)B;
  }
}

// ---------- kernel 6: scatter per-point outputs ----------
__global__ void k_scatter(const int* __restrict__ M,
                          const float* __restrict__ bs,
                          const float* __restrict__ sumpl,
                          const float* __restrict__ lvrepk,
                          int K, int Mmax,
                          float* __restrict__ out_pl,
                          float* __restrict__ out_lvr) {
  long long i = (long long)blockIdx.x * blockDim.x + threadIdx.x;
  if (i >= (long long)K * Mmax) return;
  int k = (int)(i / Mmax);
  int idx = M[i];
  if (idx >= 0) {
    out_pl[idx]  = (bs[idx] - Q_MIN_C) / (sumpl[k] + 1e-12f);
    out_lvr[idx] = lvrepk[k];
  }
}

// ---------- host launcher ----------
extern "C" void kernel_launch(void* const* d_in, const int* in_sizes, int n_in,
                              void* d_out, int out_size, void* d_ws, size_t ws_size,
                              hipStream_t stream) {
  const float* beta   = (const float*)d_in[0];
  const float* coords = (const float*)d_in[1];
  const int*   asso   = (const int*)d_in[2];
  const int*   M      = (const int*)d_in[3];
  // d_in[4] (M_not) is intentionally unused: dense-minus-members restructuring.
  const int*   kper   = (const int*)d_in[5];
  const int*   rowi   = (const int*)d_in[6];
  const int*   lengths= (const int*)d_in[7];

  int N    = in_sizes[0];
  int C    = in_sizes[1] / N;
  int K    = in_sizes[5];
  int Mmax = in_sizes[3] / K;
  int B    = in_sizes[7];
  int Nper = N / B;
  int Kper = K / B;
  int KT   = (Kper + 15) / 16;

  float* w      = (float*)d_ws;
  float* bs     = w;                 // N
  float* nbuf   = bs + N;            // N
  float* xa     = nbuf + N;          // K*C
  float* na     = xa + (size_t)K*C;  // K
  float* bsk    = na + K;            // K
  float* cntf   = bsk + K;           // K
  float* sumpl  = cntf + K;          // K
  float* LVk    = sumpl + K;         // K
  float* pen    = LVk + K;           // K
  float* repcorr= pen + K;           // K
  float* lrepraw= repcorr + K;       // K  (atomic accumulator, zeroed)
  float* noise  = lrepraw + K;       // B  (atomic accumulator, zeroed)
  float* lvrepk = noise + B;         // K

  float* out    = (float*)d_out;
  float* out_pl = out + 3;
  float* out_lvr= out + 3 + N;

  int zn = 2 * N;
  int zmax = zn > (K + B) ? zn : (K + B);
  k_zero<<<(zmax + 255) / 256, 256, 0, stream>>>(out_pl, zn, lrepraw, K + B);

  k_point<<<(N + 255) / 256, 256, 0, stream>>>(beta, coords, N, C, bs, nbuf);

  k_cluster<<<K, 256, 0, stream>>>(beta, coords, M, bs, C, Mmax,
                                   xa, na, bsk, cntf, sumpl, LVk, pen, repcorr);

  k_noise<<<(N + 255) / 256, 256, 0, stream>>>(beta, asso, rowi, N, noise);

  int gx = (Nper / 16 + (8 * 2) - 1) / (8 * 2);   // ~2 chunks per wave
  if (gx < 1) gx = 1;
  if (gx > 128) gx = 128;
  dim3 grep(gx, B * KT);
  k_rep<<<grep, 256, 0, stream>>>(coords, bs, nbuf, xa, na, bsk,
                                  C, Kper, Nper, KT, lrepraw);

  k_final<<<1, 256, 0, stream>>>(LVk, pen, repcorr, lrepraw, cntf, kper,
                                 noise, lengths, K, B, Nper, out, lvrepk);

  long long tot = (long long)K * Mmax;
  k_scatter<<<(int)((tot + 255) / 256), 256, 0, stream>>>(M, bs, sumpl, lvrepk,
                                                          K, Mmax, out_pl, out_lvr);
}